// GEMConv_62689342652486
// MI455X (gfx1250) — compile-verified
//
#include <hip/hip_runtime.h>

typedef __attribute__((ext_vector_type(2))) float v2f;
typedef __attribute__((ext_vector_type(8))) float v8f;

#define V_NODES   25000
#define E_EDGES   400000
#define DIM_IN    16
#define DIM_OUT   32
#define NUM_BASIS 5
#define K_TOT     80            // NUM_BASIS * DIM_IN
#define G_STRIDE  84            // padded g-row stride (dwords, even => b64 aligned)
#define WB_OSTR   18            // padded W_basis per-output-row stride (even)
#define WB_KSTR   (32 * WB_OSTR)

// ---------------------------------------------------------------------------
// Kernel 1: out[v, :] = x[v, :] @ W_self^T   (initializes the output buffer)
// One wave = one 16-node tile, N=32 as two 16-wide WMMA accumulators, K=16.
// ---------------------------------------------------------------------------
__global__ __launch_bounds__(256)
void gem_self_kernel(const float* __restrict__ x,
                     const float* __restrict__ W_self,   // [32][16]
                     float* __restrict__ out)            // [V][32]
{
    const int lane = threadIdx.x & 31;
    const int wave = threadIdx.x >> 5;
    const int m    = lane & 15;
    const int hi   = lane >> 4;
    const int base = (blockIdx.x * 8 + wave) * 16;
    if (base >= V_NODES) return;                 // wave-uniform branch

    int nodeA = base + m;
    if (nodeA >= V_NODES) nodeA = V_NODES - 1;   // clamp loads; stores guarded

    v8f c0 = {}; v8f c1 = {};
#pragma unroll
    for (int kc = 0; kc < 4; ++kc) {
        const int k0 = kc * 4 + hi * 2;          // even => 8B-aligned b64 loads
        v2f a  = *(const v2f*)(x + nodeA * DIM_IN + k0);
        v2f b0 = *(const v2f*)(W_self + m * DIM_IN + k0);
        v2f b1 = *(const v2f*)(W_self + (m + 16) * DIM_IN + k0);
        c0 = __builtin_amdgcn_wmma_f32_16x16x4_f32(false, a, false, b0,
                                                   (short)0, c0, false, false);
        c1 = __builtin_amdgcn_wmma_f32_16x16x4_f32(false, a, false, b1,
                                                   (short)0, c1, false, false);
    }

    if (base + 16 <= V_NODES) {
        // Fast path (all tiles but the last): straight-line stores, no exec juggling.
#pragma unroll
        for (int j = 0; j < 8; ++j) {
            const int row = base + j + 8 * hi;   // D: VGPR j -> M = j + 8*hi
            out[row * DIM_OUT + m]      = c0[j];
            out[row * DIM_OUT + m + 16] = c1[j];
        }
    } else {
#pragma unroll
        for (int j = 0; j < 8; ++j) {
            const int row = base + j + 8 * hi;
            if (row < V_NODES) {
                out[row * DIM_OUT + m]      = c0[j];
                out[row * DIM_OUT + m + 16] = c1[j];
            }
        }
    }
}

// ---------------------------------------------------------------------------
// Kernel 2: per-edge messages as GEMM  (16 edges x K=80) @ (80 x 32), then
// atomic scatter-add into out[tgt]. One wave per 16-edge tile.
// ---------------------------------------------------------------------------
__global__ __launch_bounds__(256)
void gem_edge_kernel(const float* __restrict__ x,
                     const long long* __restrict__ edge_index, // [2][E] int64
                     const float* __restrict__ angles,         // [E]
                     const float* __restrict__ transporters,   // [E]
                     const float* __restrict__ W_basis,        // [5][32][16]
                     float* __restrict__ out)                  // [V][32]
{
    __shared__ float sWb[NUM_BASIS * WB_KSTR];     // padded W_basis (2880 f)
    __shared__ float sG[8 * 16 * G_STRIDE];        // per-wave g tiles (10752 f)

    const int tid  = threadIdx.x;
    const int lane = tid & 31;
    const int wave = tid >> 5;
    const int m    = lane & 15;
    const int hi   = lane >> 4;
    const int e0   = (blockIdx.x * 8 + wave) * 16; // E divides exactly

    // Stage W_basis into bank-padded LDS with async global->LDS DMA
    // (GLOBAL_LOAD_ASYNC_TO_LDS_B32, tracked by ASYNCcnt; no VGPR round trip).
    // 2560 elements / 256 threads = 10 per thread, exact.
#pragma unroll
    for (int it = 0; it < 10; ++it) {
        const int idx = tid + it * 256;
        const int kb  = idx >> 9;          // / 512
        const int r   = idx & 511;
        const int o   = r >> 4;
        const int i   = r & 15;
        // Flat LDS addresses carry the LDS byte offset in their low 32 bits.
        const unsigned           ldsoff = (unsigned)(uintptr_t)(&sWb[kb * WB_KSTR + o * WB_OSTR + i]);
        const unsigned long long gaddr  = (unsigned long long)(uintptr_t)(W_basis + idx);
        asm volatile("global_load_async_to_lds_b32 %0, %1, off"
                     :: "v"(ldsoff), "v"(gaddr) : "memory");
    }

    // Lanes 0..15: gather, parallel-transport, expand with Fourier basis.
    // (tgt for this tile also loaded here, broadcast later via lane permute)
    const int tgt_m = (int)edge_index[E_EDGES + e0 + m];   // all lanes; hi dup

    if (lane < 16) {
        const int  e = e0 + lane;
        const int  s = (int)edge_index[e];
        const float t = transporters[e];
        const float a = angles[e];

        float f[DIM_IN];
#pragma unroll
        for (int i = 0; i < DIM_IN; ++i) f[i] = x[s * DIM_IN + i];

        const float ct = cosf(t), st = sinf(t);
#pragma unroll
        for (int p = 0; p < 4; ++p) {
            const float v0 = f[8 + 2 * p];
            const float v1 = f[8 + 2 * p + 1];
            f[8 + 2 * p]     = v0 * ct - v1 * st;
            f[8 + 2 * p + 1] = v0 * st + v1 * ct;
        }

        const float ca = cosf(a), sa = sinf(a);
        float bas[NUM_BASIS];
        bas[0] = 1.0f;
        bas[1] = ca;
        bas[2] = sa;
        bas[3] = ca * ca - sa * sa;   // cos 2a
        bas[4] = 2.0f * sa * ca;      // sin 2a

        float* gr = &sG[(wave * 16 + lane) * G_STRIDE];
#pragma unroll
        for (int k = 0; k < NUM_BASIS; ++k)
#pragma unroll
            for (int i = 0; i < DIM_IN; ++i)
                gr[k * 16 + i] = bas[k] * f[i];
    }

    // Wait for this wave's async LDS fills, then block barrier so every wave
    // sees the complete sWb staging (each wave waited on its own ASYNCcnt).
    asm volatile("s_wait_asynccnt 0" ::: "memory");
    __syncthreads();

    // K = 80 in 20 chunks of 4; two N-tiles (cols 0..15 and 16..31).
    v8f c0 = {}; v8f c1 = {};
    const float* gbase = &sG[wave * 16 * G_STRIDE];
#pragma unroll
    for (int kc = 0; kc < 20; ++kc) {
        const int k0 = kc * 4 + hi * 2;          // even offsets
        v2f a = *(const v2f*)(gbase + m * G_STRIDE + k0);
        const int kb = k0 >> 4;
        const int i0 = k0 & 15;
        v2f b0 = *(const v2f*)(sWb + kb * WB_KSTR + m * WB_OSTR + i0);
        v2f b1 = *(const v2f*)(sWb + kb * WB_KSTR + (m + 16) * WB_OSTR + i0);
        c0 = __builtin_amdgcn_wmma_f32_16x16x4_f32(false, a, false, b0,
                                                   (short)0, c0, false, false);
        c1 = __builtin_amdgcn_wmma_f32_16x16x4_f32(false, a, false, b1,
                                                   (short)0, c1, false, false);
    }

    // Scatter-add messages into out[tgt]; tgt broadcast via lane permute.
#pragma unroll
    for (int j = 0; j < 8; ++j) {
        const int t = __shfl(tgt_m, j + 8 * hi, 32);   // D row -> edge j+8*hi
        atomicAdd(&out[t * DIM_OUT + m],      c0[j]);
        atomicAdd(&out[t * DIM_OUT + m + 16], c1[j]);
    }
}

// ---------------------------------------------------------------------------
extern "C" void kernel_launch(void* const* d_in, const int* in_sizes, int n_in,
                              void* d_out, int out_size, void* d_ws, size_t ws_size,
                              hipStream_t stream) {
    const float*     x            = (const float*)d_in[0];
    const long long* edge_index   = (const long long*)d_in[1];
    const float*     angles       = (const float*)d_in[2];
    const float*     transporters = (const float*)d_in[3];
    const float*     W_self       = (const float*)d_in[4];
    const float*     W_basis      = (const float*)d_in[5];
    float*           out          = (float*)d_out;

    // Kernel 1: 1563 node tiles, 8 waves/block -> 196 blocks. Writes every
    // element of out (harness poisons d_out), so no memset needed.
    const int node_tiles = (V_NODES + 15) / 16;
    const int grid1 = (node_tiles + 7) / 8;
    gem_self_kernel<<<grid1, 256, 0, stream>>>(x, W_self, out);

    // Kernel 2: 25000 edge tiles / 8 waves per block = 3125 blocks.
    const int grid2 = (E_EDGES / 16) / 8;
    gem_edge_kernel<<<grid2, 256, 0, stream>>>(x, edge_index, angles,
                                               transporters, W_basis, out);
}